// ProbAttention_10316511445593
// MI455X (gfx1250) — compile-verified
//
#include <hip/hip_runtime.h>
#include <hip/hip_bf16.h>
#include <math.h>

typedef __attribute__((ext_vector_type(2))) float v2f;
typedef __attribute__((ext_vector_type(8))) float v8f;

#define B_   4
#define L_   4096
#define H_   8
#define D_   64
#define U_   45
#define SK_  45
#define NCH_ 64      // cumsum chunks per (b,h)
#define CHL_ 64      // L elements per chunk

#define NEG_BIG (-3.0e38f)   // finite -inf sentinel (exp(NEG_BIG - x) underflows to 0)

__device__ __forceinline__ v8f vzero8() {
  v8f r;
  #pragma unroll
  for (int i = 0; i < 8; ++i) r[i] = 0.f;
  return r;
}

// Async global->LDS copy of one 16-byte packet (CDNA5 GLOBAL_LOAD_ASYNC_TO_LDS_B128).
// ldsoff: per-lane LDS byte offset; gaddr: per-lane global address.
__device__ __forceinline__ void async_copy_b128(unsigned ldsoff, const void* gaddr) {
  asm volatile("global_load_async_to_lds_b128 %0, %1, off"
               :: "v"(ldsoff), "v"(gaddr) : "memory");
}
__device__ __forceinline__ void wait_asynccnt0() {
  asm volatile("s_wait_asynccnt 0x0" ::: "memory");
}

// ---------------------------------------------------------------------------
// Kernel 1: sparsity measure  M[b,h,q] = max_s(QK_s) - sum_s(QK_s)/L
// inputs are (B, L, H, D) layout
// ---------------------------------------------------------------------------
__global__ __launch_bounds__(256) void measure_kernel(
    const float* __restrict__ Q, const float* __restrict__ K,
    const int* __restrict__ idxS, float* __restrict__ M) {
  int tid = blockIdx.x * blockDim.x + threadIdx.x;
  if (tid >= B_ * H_ * L_) return;
  int q  = tid % L_;
  int bh = tid / L_;
  int h  = bh % H_;
  int b  = bh / H_;

  const float4* Qrow = (const float4*)(Q + ((size_t)(b * L_ + q) * H_ + h) * D_);
  float4 qv[16];
  #pragma unroll
  for (int i = 0; i < 16; ++i) qv[i] = Qrow[i];

  float mx = -INFINITY, sm = 0.f;
  for (int s = 0; s < SK_; ++s) {
    int kidx = idxS[q * SK_ + s];
    const float4* Krow = (const float4*)(K + ((size_t)(b * L_ + kidx) * H_ + h) * D_);
    float dot = 0.f;
    #pragma unroll
    for (int i = 0; i < 16; ++i) {
      float4 kv = Krow[i];
      dot += qv[i].x * kv.x + qv[i].y * kv.y + qv[i].z * kv.z + qv[i].w * kv.w;
    }
    mx = fmaxf(mx, dot);
    sm += dot;
  }
  M[tid] = mx - sm * (1.0f / (float)L_);
}

// ---------------------------------------------------------------------------
// Kernel 2: top-45 indices of M per (b,h).  One block per (b,h).
// ---------------------------------------------------------------------------
__global__ __launch_bounds__(256) void topk_kernel(
    const float* __restrict__ M, int* __restrict__ Mtop) {
  int bh = blockIdx.x;
  int t  = threadIdx.x;
  __shared__ float vals[L_];
  __shared__ float redV[256];
  __shared__ int   redI[256];

  for (int i = t; i < L_; i += 256) vals[i] = M[(size_t)bh * L_ + i];
  __syncthreads();

  for (int k = 0; k < U_; ++k) {
    float bv = -INFINITY;
    int   bi = 0x7fffffff;
    for (int i = t; i < L_; i += 256) {
      float v = vals[i];
      if (v > bv || (v == bv && i < bi)) { bv = v; bi = i; }
    }
    redV[t] = bv; redI[t] = bi;
    __syncthreads();
    for (int s = 128; s > 0; s >>= 1) {
      if (t < s) {
        float v2 = redV[t + s]; int i2 = redI[t + s];
        if (v2 > redV[t] || (v2 == redV[t] && i2 < redI[t])) {
          redV[t] = v2; redI[t] = i2;
        }
      }
      __syncthreads();
    }
    if (t == 0) {
      Mtop[bh * U_ + k] = redI[0];
      vals[redI[0]] = -INFINITY;
    }
    __syncthreads();
  }
}

// ---------------------------------------------------------------------------
// Kernels 3a/3b: cumsum of V along L into output (B,H,L,D) layout
// ---------------------------------------------------------------------------
__global__ __launch_bounds__(64) void csum1_kernel(
    const float* __restrict__ V, float* __restrict__ chunksum) {
  int bh = blockIdx.x / NCH_;
  int ch = blockIdx.x % NCH_;
  int b  = bh / H_, h = bh % H_;
  int d  = threadIdx.x;
  float s = 0.f;
  for (int i = 0; i < CHL_; ++i) {
    int l = ch * CHL_ + i;
    s += V[((size_t)(b * L_ + l) * H_ + h) * D_ + d];
  }
  chunksum[((size_t)bh * NCH_ + ch) * D_ + d] = s;
}

__global__ __launch_bounds__(64) void csum2_kernel(
    const float* __restrict__ V, const float* __restrict__ chunksum,
    float* __restrict__ ctx) {
  int bh = blockIdx.x / NCH_;
  int ch = blockIdx.x % NCH_;
  int b  = bh / H_, h = bh % H_;
  int d  = threadIdx.x;
  float run = 0.f;
  for (int c = 0; c < ch; ++c)
    run += chunksum[((size_t)bh * NCH_ + c) * D_ + d];
  for (int i = 0; i < CHL_; ++i) {
    int l = ch * CHL_ + i;
    run += V[((size_t)(b * L_ + l) * H_ + h) * D_ + d];
    ctx[((size_t)bh * L_ + l) * D_ + d] = run;
  }
}

// ---------------------------------------------------------------------------
// Kernel 4: WMMA attention for the 45 selected queries per (b,h), scattered
// into ctx.  One wave (32 lanes) per (b,h, 16-query tile).  3 tiles cover 45.
// V_WMMA_F32_16X16X4_F32 + async global->LDS staging; two-pass softmax with
// branchless per-lane online stats.
// ---------------------------------------------------------------------------
__global__ __launch_bounds__(32) void attn_kernel(
    const float* __restrict__ Q, const float* __restrict__ K,
    const float* __restrict__ V, const int* __restrict__ Mtop,
    float* __restrict__ ctx) {
  const float scale = 0.125f;  // 1/sqrt(64)
  int bh = blockIdx.x / 3;
  int t  = blockIdx.x % 3;
  int b  = bh / H_, h = bh % H_;
  int lane = threadIdx.x;
  int half = lane >> 4;      // 0: lanes 0-15, 1: lanes 16-31
  int lm   = lane & 15;

  __shared__ __align__(16) float ldsK[16 * D_];
  __shared__ __align__(16) float ldsV[16 * D_];
  __shared__ __align__(16) float ldsP[16 * 16];
  __shared__ int qposS[16];

  if (lane < 16) {
    int ut = t * 16 + lane;
    qposS[lane] = (ut < U_) ? Mtop[bh * U_ + ut] : (L_ - 1);
  }
  __syncthreads();

  int rowpos[8];
  #pragma unroll
  for (int j = 0; j < 8; ++j) rowpos[j] = qposS[j + 8 * half];
  int aqpos = qposS[lm];

  int tileMaxPos = 0;
  #pragma unroll
  for (int i = 0; i < 16; ++i) tileMaxPos = max(tileMaxPos, qposS[i]);
  int ktEnd = (tileMaxPos >> 4) + 1;   // exclusive

  // per-lane async-staging coordinates: lane copies 8 x b128 packets per tile
  // packet e = i*32+lane -> row r = e>>4, float4-column c4 = (e&15)*4
  unsigned kLds[8], vLds[8];
  size_t   rowByteOff[8];
  #pragma unroll
  for (int i = 0; i < 8; ++i) {
    int e = i * 32 + lane;
    int r = e >> 4, c4 = (e & 15) * 4;
    kLds[i] = (unsigned)(unsigned long long)&ldsK[r * D_ + c4];
    vLds[i] = (unsigned)(unsigned long long)&ldsV[r * D_ + c4];
    rowByteOff[i] = ((size_t)r * H_ * D_ + c4) * sizeof(float);
  }

  // A-fragments of Q (16x4 f32 layout): lane holds Q[row=lm][4c + 2*half + {0,1}]
  v2f aq[16];
  {
    const float* Qrow = Q + ((size_t)(b * L_ + aqpos) * H_ + h) * D_;
    #pragma unroll
    for (int c = 0; c < 16; ++c)
      aq[c] = *(const v2f*)(Qrow + 4 * c + 2 * half);
  }

  // -------- Phase A: row max + sum-of-exp (branchless online, then merge) ---
  float mrow[8], lrow[8];
  #pragma unroll
  for (int j = 0; j < 8; ++j) { mrow[j] = NEG_BIG; lrow[j] = 0.f; }

  for (int kt = 0; kt < ktEnd; ++kt) {
    __syncthreads();
    {
      const char* Kbase =
          (const char*)(K + ((size_t)(b * L_ + kt * 16) * H_ + h) * D_);
      #pragma unroll
      for (int i = 0; i < 8; ++i)
        async_copy_b128(kLds[i], Kbase + rowByteOff[i]);
    }
    wait_asynccnt0();
    __syncthreads();

    v8f acc = vzero8();
    #pragma unroll
    for (int c = 0; c < 16; ++c) {
      v2f bf = *(const v2f*)&ldsK[lm * D_ + 4 * c + 2 * half];
      acc = __builtin_amdgcn_wmma_f32_16x16x4_f32(
          false, aq[c], false, bf, (short)0, acc, false, false);
    }

    int keyg = kt * 16 + lm;
    #pragma unroll
    for (int j = 0; j < 8; ++j) {
      float v  = (keyg <= rowpos[j]) ? acc[j] * scale : NEG_BIG;
      float nm = fmaxf(mrow[j], v);
      lrow[j]  = lrow[j] * __expf(mrow[j] - nm) + __expf(v - nm);
      mrow[j]  = nm;
    }
  }

  // merge softmax stats across the 16 lanes of each half (branchless)
  float rl[8];
  #pragma unroll
  for (int j = 0; j < 8; ++j) {
    float m = mrow[j], l = lrow[j];
    for (int s = 1; s < 16; s <<= 1) {
      float m2 = __shfl_xor(m, s, 32);
      float l2 = __shfl_xor(l, s, 32);
      float mm = fmaxf(m, m2);
      l = l * __expf(m - mm) + l2 * __expf(m2 - mm);
      m = mm;
    }
    mrow[j] = m;
    rl[j]   = 1.0f / l;   // l >= 1: the row max contributes exp(0)=1
  }

  // -------- Phase B: recompute S, form P, accumulate O = P @ V --------------
  v8f o[4];
  #pragma unroll
  for (int dt = 0; dt < 4; ++dt) o[dt] = vzero8();

  for (int kt = 0; kt < ktEnd; ++kt) {
    __syncthreads();
    {
      const char* Kbase =
          (const char*)(K + ((size_t)(b * L_ + kt * 16) * H_ + h) * D_);
      const char* Vbase =
          (const char*)(V + ((size_t)(b * L_ + kt * 16) * H_ + h) * D_);
      #pragma unroll
      for (int i = 0; i < 8; ++i) {
        async_copy_b128(kLds[i], Kbase + rowByteOff[i]);
        async_copy_b128(vLds[i], Vbase + rowByteOff[i]);
      }
    }
    wait_asynccnt0();
    __syncthreads();

    v8f acc = vzero8();
    #pragma unroll
    for (int c = 0; c < 16; ++c) {
      v2f bf = *(const v2f*)&ldsK[lm * D_ + 4 * c + 2 * half];
      acc = __builtin_amdgcn_wmma_f32_16x16x4_f32(
          false, aq[c], false, bf, (short)0, acc, false, false);
    }

    int keyg = kt * 16 + lm;
    #pragma unroll
    for (int j = 0; j < 8; ++j) {
      float p = 0.f;
      if (keyg <= rowpos[j])
        p = __expf(acc[j] * scale - mrow[j]) * rl[j];
      ldsP[(j + 8 * half) * 16 + lm] = p;   // P row-major [query][key]
    }
    __syncthreads();

    #pragma unroll
    for (int dt = 0; dt < 4; ++dt) {
      #pragma unroll
      for (int k0 = 0; k0 < 16; k0 += 4) {
        v2f af = *(const v2f*)&ldsP[lm * 16 + k0 + 2 * half];
        v2f bf;
        bf.x = ldsV[(k0 + 2 * half)     * D_ + dt * 16 + lm];
        bf.y = ldsV[(k0 + 1 + 2 * half) * D_ + dt * 16 + lm];
        o[dt] = __builtin_amdgcn_wmma_f32_16x16x4_f32(
            false, af, false, bf, (short)0, o[dt], false, false);
      }
    }
  }

  // -------- scatter O rows into ctx at the top-k positions ------------------
  #pragma unroll
  for (int j = 0; j < 8; ++j) {
    int row = j + 8 * half;
    if (t * 16 + row < U_) {
      size_t base = ((size_t)bh * L_ + rowpos[j]) * D_ + lm;
      ctx[base +  0] = o[0][j];
      ctx[base + 16] = o[1][j];
      ctx[base + 32] = o[2][j];
      ctx[base + 48] = o[3][j];
    }
  }
}

// ---------------------------------------------------------------------------
extern "C" void kernel_launch(void* const* d_in, const int* in_sizes, int n_in,
                              void* d_out, int out_size, void* d_ws, size_t ws_size,
                              hipStream_t stream) {
  const float* Q    = (const float*)d_in[0];
  const float* K    = (const float*)d_in[1];
  const float* V    = (const float*)d_in[2];
  const int*   idxS = (const int*)d_in[3];
  float* ctx = (float*)d_out;

  // workspace layout
  float* Mbuf     = (float*)d_ws;                                       // 131072 f
  int*   Mtop     = (int*)((char*)d_ws + (size_t)B_ * H_ * L_ * 4);     // 1440 i
  float* chunksum = (float*)((char*)d_ws + (size_t)B_ * H_ * L_ * 4 + 8192);

  {
    int total = B_ * H_ * L_;
    measure_kernel<<<(total + 255) / 256, 256, 0, stream>>>(Q, K, idxS, Mbuf);
  }
  topk_kernel<<<B_ * H_, 256, 0, stream>>>(Mbuf, Mtop);
  csum1_kernel<<<B_ * H_ * NCH_, 64, 0, stream>>>(V, chunksum);
  csum2_kernel<<<B_ * H_ * NCH_, 64, 0, stream>>>(V, chunksum, ctx);
  attn_kernel<<<B_ * H_ * 3, 32, 0, stream>>>(Q, K, V, Mtop, ctx);
}